// GNN_2619930050604
// MI455X (gfx1250) — compile-verified
//
#include <hip/hip_runtime.h>
#include <hip/hip_bf16.h>

#define NNODES 100000
#define NEDGES 1600000
#define NGRAPH 64
#define CH 128
#define NCLS 8

typedef __attribute__((ext_vector_type(2))) float v2f;
typedef __attribute__((ext_vector_type(8))) float v8f;

__device__ __forceinline__ void atomAddF(float* p, float v) {
    // lowers to global_atomic_add_f32 (no CAS loop)
    unsafeAtomicAdd(p, v);
}

// ---------------- degree / norm ----------------
__global__ void k_init_deg(int* deg) {
    int i = blockIdx.x * blockDim.x + threadIdx.x;
    if (i < NNODES) deg[i] = 1;   // self-loop
}

__global__ void k_deg(const int* __restrict__ dst, int* deg) {
    int e = blockIdx.x * blockDim.x + threadIdx.x;
    if (e < NEDGES) atomicAdd(&deg[dst[e]], 1);
}

__global__ void k_dinv(const int* __restrict__ deg, float* __restrict__ dinv) {
    int i = blockIdx.x * blockDim.x + threadIdx.x;
    if (i < NNODES) dinv[i] = rsqrtf((float)deg[i]);
}

__global__ void k_zero(float* __restrict__ p, int n) {
    int i = blockIdx.x * blockDim.x + threadIdx.x;
    if (i < n) p[i] = 0.0f;
}

// ---------------- WMMA fp32 GEMM: Y[N,128] = X[N,128] @ W[128,128] ----------------
// One wave computes a 16x16 output tile via 32x V_WMMA_F32_16X16X4_F32.
// 8 waves/block -> 16 rows x 128 cols per block. N % 16 == 0, no guards: EXEC all-1s.
__global__ __launch_bounds__(256) void k_gemm_wmma(const float* __restrict__ X,
                                                   const float* __restrict__ W,
                                                   float* __restrict__ Y) {
    const int row0 = blockIdx.x << 4;
    const int wave = threadIdx.x >> 5;
    const int lane = threadIdx.x & 31;
    const int col0 = wave << 4;
    const int mn   = lane & 15;   // M for A-loads, N for B-loads/stores
    const int half = lane >> 4;   // K-half select

    // A 16x4 f32: lanes 0-15 hold K=0,1 ; lanes 16-31 hold K=2,3 (VGPR0/1)
    const float* xr = X + (size_t)(row0 + mn) * CH + (half << 1);
    // B 4x16 f32: VGPR v -> K = v + 2*half, N = mn
    const float* wp = W + (size_t)(half << 1) * CH + col0 + mn;

    v8f c = {};
#pragma unroll
    for (int k = 0; k < CH; k += 4) {
        v2f a = *(const v2f*)(xr + k);
        v2f b;
        b[0] = wp[(size_t)k * CH];
        b[1] = wp[(size_t)(k + 1) * CH];
        c = __builtin_amdgcn_wmma_f32_16x16x4_f32(false, a, false, b,
                                                  (short)0, c, false, false);
    }

    // C/D 16x16 f32: VGPR r -> M = r + 8*half, N = mn
    float* yr = Y + (size_t)row0 * CH + col0 + mn;
#pragma unroll
    for (int r = 0; r < 8; ++r)
        yr[(size_t)(r + (half << 3)) * CH] = c[r];
}

// ---------------- edge scatter: agg[dst] += xw[src] * dinv[src]*dinv[dst] ----------------
// One wave per edge; each lane moves float4 (global_load_b128 + 4x global_atomic_add_f32).
__global__ __launch_bounds__(256) void k_scatter(const int* __restrict__ src,
                                                 const int* __restrict__ dst,
                                                 const float* __restrict__ dinv,
                                                 const float* __restrict__ xw,
                                                 float* __restrict__ agg) {
    const int e = (blockIdx.x << 3) + (threadIdx.x >> 5);
    if (e >= NEDGES) return;
    const int lane = threadIdx.x & 31;
    const int s = src[e];
    const int d = dst[e];
    const float nrm = dinv[s] * dinv[d];
    const float4 v = *(const float4*)(xw + (size_t)s * CH + (lane << 2));
    float* o = agg + (size_t)d * CH + (lane << 2);
    atomAddF(o + 0, v.x * nrm);
    atomAddF(o + 1, v.y * nrm);
    atomAddF(o + 2, v.z * nrm);
    atomAddF(o + 3, v.w * nrm);
}

// ---------------- epilogue layer 1: h = relu(agg + xw*dinv^2 + b1) ----------------
__global__ void k_finish_relu(const float* __restrict__ agg, const float* __restrict__ xw,
                              const float* __restrict__ dinv, const float* __restrict__ bias,
                              float* __restrict__ h) {
    const int t = blockIdx.x * blockDim.x + threadIdx.x;
    const int i = t >> 5;
    if (i >= NNODES) return;
    const int c = (t & 31) << 2;
    const float self = dinv[i] * dinv[i];
    const float4 a  = *(const float4*)(agg + (size_t)i * CH + c);
    const float4 x  = *(const float4*)(xw  + (size_t)i * CH + c);
    const float4 bb = *(const float4*)(bias + c);
    float4 r;
    r.x = fmaxf(fmaf(x.x, self, a.x) + bb.x, 0.0f);
    r.y = fmaxf(fmaf(x.y, self, a.y) + bb.y, 0.0f);
    r.z = fmaxf(fmaf(x.z, self, a.z) + bb.z, 0.0f);
    r.w = fmaxf(fmaf(x.w, self, a.w) + bb.w, 0.0f);
    *(float4*)(h + (size_t)i * CH + c) = r;
}

// ---------------- epilogue layer 2 + mean-pool accumulation ----------------
__global__ void k_finish_pool(const float* __restrict__ agg, const float* __restrict__ hw,
                              const float* __restrict__ dinv, const float* __restrict__ bias,
                              const int* __restrict__ batch,
                              float* __restrict__ poolsum, float* __restrict__ cnt) {
    const int t = blockIdx.x * blockDim.x + threadIdx.x;
    const int i = t >> 5;
    if (i >= NNODES) return;
    const int lane = t & 31;
    const int c = lane << 2;
    const int g = batch[i];
    const float self = dinv[i] * dinv[i];
    const float4 a  = *(const float4*)(agg + (size_t)i * CH + c);
    const float4 x  = *(const float4*)(hw  + (size_t)i * CH + c);
    const float4 bb = *(const float4*)(bias + c);
    float* ps = poolsum + (size_t)g * CH + c;
    atomAddF(ps + 0, fmaf(x.x, self, a.x) + bb.x);
    atomAddF(ps + 1, fmaf(x.y, self, a.y) + bb.y);
    atomAddF(ps + 2, fmaf(x.z, self, a.z) + bb.z);
    atomAddF(ps + 3, fmaf(x.w, self, a.w) + bb.w);
    if (lane == 0) atomAddF(&cnt[g], 1.0f);
}

// ---------------- head: out[64,8] = (poolsum/cnt) @ lin_W + lin_b ----------------
__global__ void k_head(const float* __restrict__ poolsum, const float* __restrict__ cnt,
                       const float* __restrict__ linW, const float* __restrict__ linb,
                       float* __restrict__ out) {
    const int t = threadIdx.x;           // 512 = 64 graphs * 8 classes
    const int g = t >> 3, n = t & 7;
    const float inv = 1.0f / fmaxf(cnt[g], 1.0f);
    float acc = linb[n];
    for (int c = 0; c < CH; ++c)
        acc = fmaf(poolsum[g * CH + c] * inv, linW[c * NCLS + n], acc);
    out[g * NCLS + n] = acc;
}

extern "C" void kernel_launch(void* const* d_in, const int* in_sizes, int n_in,
                              void* d_out, int out_size, void* d_ws, size_t ws_size,
                              hipStream_t stream) {
    const float* x    = (const float*)d_in[0];
    const int*   eidx = (const int*)  d_in[1];   // [2, E] row-major
    const int*   batch= (const int*)  d_in[2];
    const float* W1   = (const float*)d_in[3];
    const float* b1   = (const float*)d_in[4];
    const float* W2   = (const float*)d_in[5];
    const float* b2   = (const float*)d_in[6];
    const float* linW = (const float*)d_in[7];
    const float* linb = (const float*)d_in[8];
    float* out = (float*)d_out;

    const int* src = eidx;
    const int* dst = eidx + NEDGES;

    // workspace layout
    char* ws = (char*)d_ws;
    const size_t featBytes = (size_t)NNODES * CH * sizeof(float);   // 51.2 MB
    float* bufA   = (float*)ws; ws += featBytes;                    // xw / hw
    float* bufB   = (float*)ws; ws += featBytes;                    // agg
    float* bufC   = (float*)ws; ws += featBytes;                    // h (relu)
    int*   deg    = (int*)  ws; ws += (size_t)NNODES * sizeof(int);
    float* dinv   = (float*)ws; ws += (size_t)NNODES * sizeof(float);
    float* poolsum= (float*)ws; ws += (size_t)NGRAPH * CH * sizeof(float);
    float* cnt    = (float*)ws;                                     // contiguous after poolsum

    const dim3 B(256);
    const int NFEAT = NNODES * CH;                  // 12.8M

    // normalization coefficients
    k_init_deg<<<(NNODES + 255) / 256, B, 0, stream>>>(deg);
    k_deg     <<<(NEDGES + 255) / 256, B, 0, stream>>>(dst, deg);
    k_dinv    <<<(NNODES + 255) / 256, B, 0, stream>>>(deg, dinv);

    // ---- layer 1 ----
    k_gemm_wmma<<<NNODES / 16, B, 0, stream>>>(x, W1, bufA);
    k_zero     <<<(NFEAT + 255) / 256, B, 0, stream>>>(bufB, NFEAT);
    k_scatter  <<<NEDGES / 8, B, 0, stream>>>(src, dst, dinv, bufA, bufB);
    k_finish_relu<<<(NNODES * 32 + 255) / 256, B, 0, stream>>>(bufB, bufA, dinv, b1, bufC);

    // ---- layer 2 ----
    k_gemm_wmma<<<NNODES / 16, B, 0, stream>>>(bufC, W2, bufA);
    k_zero     <<<(NFEAT + 255) / 256, B, 0, stream>>>(bufB, NFEAT);
    k_scatter  <<<NEDGES / 8, B, 0, stream>>>(src, dst, dinv, bufA, bufB);

    // ---- pool + head ----
    k_zero<<<(NGRAPH * CH + NGRAPH + 255) / 256, B, 0, stream>>>(poolsum, NGRAPH * CH + NGRAPH);
    k_finish_pool<<<(NNODES * 32 + 255) / 256, B, 0, stream>>>(bufB, bufA, dinv, b2, batch,
                                                               poolsum, cnt);
    k_head<<<1, 512, 0, stream>>>(poolsum, cnt, linW, linb, out);
}